// InverseLoss_28071906247271
// MI455X (gfx1250) — compile-verified
//
#include <hip/hip_runtime.h>
#include <math.h>

// ---------------------------------------------------------------------------
// Problem geometry
// ---------------------------------------------------------------------------
constexpr int kH = 2048, kW = 2048;          // input field
constexpr int kOH = kH - 2, kOW = kW - 2;    // VALID conv output 2046x2046
constexpr long long kN = (long long)kH * kW; // 4,194,304 pixels

// Tiling: 64x16 outputs per block, 256 threads (8 wave32 waves)
constexpr int kTW = 64, kTH = 16;
constexpr int kSW = kTW + 2;                 // 66 halo cols
constexpr int kSH = kTH + 2;                 // 18 halo rows
constexpr int kSWP = 68;                     // padded LDS row (bank-friendly)
constexpr int kSstr = kSW * 3 + 2;           // 200: strain LDS row (3 comps)
constexpr int kNStage = kSW * kSH;           // 1188 halo pixels
constexpr int kThreads = 256;
constexpr int kGX = (kOW + kTW - 1) / kTW;   // 32
constexpr int kGY = (kOH + kTH - 1) / kTH;   // 128
constexpr int kBlocks = kGX * kGY;           // 4096

// ---------------------------------------------------------------------------
// CDNA5 async global->LDS path (ASYNCcnt-tracked DMA into LDS, no VGPR data)
// Builtin parameter types (from hipcc diagnostic): param0 is a non-const
// AS1 (global) int*, so param1 is the AS3 (LDS) int* destination.
// ---------------------------------------------------------------------------
typedef __attribute__((address_space(1))) int* as1_ip;
typedef __attribute__((address_space(3))) int* as3_ip;

#if defined(__has_builtin)
#if __has_builtin(__builtin_amdgcn_global_load_async_to_lds_b32)
#define HAVE_ASYNC_LDS 1
#endif
#endif

__device__ __forceinline__ void cp_b32_to_lds(const float* g, float* l) {
#if defined(HAVE_ASYNC_LDS)
  __builtin_amdgcn_global_load_async_to_lds_b32(
      (as1_ip)const_cast<float*>(g), (as3_ip)l, /*offset=*/0, /*cpol=*/0);
#else
  *l = *g; // synchronous fallback (still single-pass through LDS)
#endif
}

__device__ __forceinline__ void wait_async_lds() {
#if defined(HAVE_ASYNC_LDS)
#if defined(__has_builtin) && __has_builtin(__builtin_amdgcn_s_wait_asynccnt)
  __builtin_amdgcn_s_wait_asynccnt(0);
#else
  asm volatile("s_wait_asynccnt 0" ::: "memory");
#endif
#endif
}

// ---------------------------------------------------------------------------
// Fused kernel: async-stage halo -> stress (LDS) -> 3x3 stencil -> partials
// ---------------------------------------------------------------------------
__global__ __launch_bounds__(kThreads) void
fused_stress_stencil(const float* __restrict__ pred_E,
                     const float* __restrict__ pred_v,
                     const float* __restrict__ strain,
                     float* __restrict__ part_f,   // [kBlocks] sum |fx/Ec|+|fy/Ec|
                     float* __restrict__ part_e)   // [kBlocks] sum pred_E
{
  __shared__ float sE [kSH][kSWP];
  __shared__ float sV [kSH][kSWP];
  __shared__ float sS [kSH][kSstr];
  __shared__ float sXX[kSH][kSWP];
  __shared__ float sYY[kSH][kSWP];
  __shared__ float sXY[kSH][kSWP];
  __shared__ float red[kThreads];

  const int t    = threadIdx.x;
  const int bx   = blockIdx.x % kGX;
  const int by   = blockIdx.x / kGX;
  const int row0 = by * kTH;
  const int col0 = bx * kTW;

  // ---- Phase 0: async DMA of the raw halo into LDS -----------------------
  for (int p = t; p < kNStage; p += kThreads) {
    const int sy = p / kSW, sx = p % kSW;
    const int gy = row0 + sy, gx = col0 + sx;
    if (gy < kH && gx < kW) {
      const int idx = gy * kW + gx;
      cp_b32_to_lds(pred_E + idx,         &sE[sy][sx]);
      cp_b32_to_lds(pred_v + idx,         &sV[sy][sx]);
      cp_b32_to_lds(strain + 3 * idx + 0, &sS[sy][3 * sx + 0]);
      cp_b32_to_lds(strain + 3 * idx + 1, &sS[sy][3 * sx + 1]);
      cp_b32_to_lds(strain + 3 * idx + 2, &sS[sy][3 * sx + 2]);
    }
  }
  wait_async_lds();   // drain this wave's ASYNCcnt
  __syncthreads();    // cross-wave visibility of LDS

  // ---- Phase 1: plane-stress constitutive law, once per halo pixel -------
  for (int p = t; p < kNStage; p += kThreads) {
    const int sy = p / kSW, sx = p % kSW;
    const float v  = sV[sy][sx];
    const float E  = sE[sy][sx];
    const float e0 = sS[sy][3 * sx + 0];
    const float e1 = sS[sy][3 * sx + 1];
    const float e2 = sS[sy][3 * sx + 2];
    const float frac = E / (1.0f - v * v);
    sXX[sy][sx] = (e0 + v * e1) * frac;
    sYY[sy][sx] = (v * e0 + e1) * frac;
    sXY[sy][sx] = (e2 * (1.0f - v) * 0.5f) * frac;
  }
  __syncthreads();

  // ---- Phase 2: 3x3 cross-correlation stencils + local accumulation ------
  // wx_xx = [-1,0,1]^T (x) [1,1,1] ; wx_xy = [1,1,1]^T (x) [1,0,-1]
  float acc = 0.0f;
  const int lx = t & (kTW - 1);
  const int ry = t >> 6;                  // 0..3
  #pragma unroll
  for (int r = 0; r < 4; ++r) {
    const int ly = ry * 4 + r;            // 0..15
    const int gi = row0 + ly, gj = col0 + lx;
    if (gi < kOH && gj < kOW) {
      const float Ec =
          sE[ly][lx]   + sE[ly][lx+1]   + sE[ly][lx+2] +
          sE[ly+1][lx] + sE[ly+1][lx+1] + sE[ly+1][lx+2] +
          sE[ly+2][lx] + sE[ly+2][lx+1] + sE[ly+2][lx+2];
      const float fx =
          (sXX[ly+2][lx] + sXX[ly+2][lx+1] + sXX[ly+2][lx+2])
        - (sXX[ly  ][lx] + sXX[ly  ][lx+1] + sXX[ly  ][lx+2])
        + (sXY[ly][lx]   + sXY[ly+1][lx]   + sXY[ly+2][lx])
        - (sXY[ly][lx+2] + sXY[ly+1][lx+2] + sXY[ly+2][lx+2]);
      const float fy =
          (sYY[ly][lx]   + sYY[ly+1][lx]   + sYY[ly+2][lx])
        - (sYY[ly][lx+2] + sYY[ly+1][lx+2] + sYY[ly+2][lx+2])
        + (sXY[ly+2][lx] + sXY[ly+2][lx+1] + sXY[ly+2][lx+2])
        - (sXY[ly  ][lx] + sXY[ly  ][lx+1] + sXY[ly  ][lx+2]);
      const float inv = 1.0f / Ec;
      acc += fabsf(fx * inv) + fabsf(fy * inv);
    }
  }

  // ---- Phase 3: grid-stride partial sum of pred_E (L2-resident re-read) --
  float accE = 0.0f;
  for (long long g = (long long)blockIdx.x * kThreads + t; g < kN;
       g += (long long)kBlocks * kThreads)
    accE += pred_E[g];

  // ---- Deterministic block reductions ------------------------------------
  __syncthreads();
  red[t] = acc;
  __syncthreads();
  for (int s = kThreads / 2; s > 0; s >>= 1) {
    if (t < s) red[t] += red[t + s];
    __syncthreads();
  }
  if (t == 0) part_f[blockIdx.x] = red[0];
  __syncthreads();
  red[t] = accE;
  __syncthreads();
  for (int s = kThreads / 2; s > 0; s >>= 1) {
    if (t < s) red[t] += red[t + s];
    __syncthreads();
  }
  if (t == 0) part_e[blockIdx.x] = red[0];
}

// ---------------------------------------------------------------------------
// Finalize: deterministic sum of 4096 partials -> scalar loss
// ---------------------------------------------------------------------------
__global__ __launch_bounds__(256) void
finalize_loss(const float* __restrict__ part_f,
              const float* __restrict__ part_e,
              float* __restrict__ out)
{
  __shared__ float rf[256];
  __shared__ float re[256];
  const int t = threadIdx.x;
  float sf = 0.0f, se = 0.0f;
  for (int i = t; i < kBlocks; i += 256) { sf += part_f[i]; se += part_e[i]; }
  rf[t] = sf; re[t] = se;
  __syncthreads();
  for (int s = 128; s > 0; s >>= 1) {
    if (t < s) { rf[t] += rf[t + s]; re[t] += re[t + s]; }
    __syncthreads();
  }
  if (t == 0) {
    const float loss_xy = rf[0] / (float)(kOH * kOW);          // mean|fx/Ec|+mean|fy/Ec|
    const float loss_e  = fabsf(re[0] / (float)kN - 1.0f);     // |mean(E) - 1|
    out[0] = loss_xy + loss_e * 0.01f;
  }
}

// ---------------------------------------------------------------------------
extern "C" void kernel_launch(void* const* d_in, const int* in_sizes, int n_in,
                              void* d_out, int out_size, void* d_ws, size_t ws_size,
                              hipStream_t stream) {
  (void)in_sizes; (void)n_in; (void)out_size; (void)ws_size;
  const float* pred_E = (const float*)d_in[0];
  const float* pred_v = (const float*)d_in[1];
  const float* strain = (const float*)d_in[2];
  float* out    = (float*)d_out;
  float* part_f = (float*)d_ws;            // kBlocks floats
  float* part_e = part_f + kBlocks;        // kBlocks floats (32 KB total)

  fused_stress_stencil<<<kBlocks, kThreads, 0, stream>>>(pred_E, pred_v, strain,
                                                         part_f, part_e);
  finalize_loss<<<1, 256, 0, stream>>>(part_f, part_e, out);
}